// UnifiedMultimodalResonator_89644557402474
// MI455X (gfx1250) — compile-verified
//
#include <hip/hip_runtime.h>
#include <hip/hip_bf16.h>

typedef __attribute__((ext_vector_type(2))) float v2f;
typedef __attribute__((ext_vector_type(4))) float v4f;
typedef __attribute__((ext_vector_type(8))) float v8f;

#define B_    64
#define DIM_  256
#define HW_   (512 * 512)
#define L_    2048
#define T_    8192
#define VCH   64            // chunks per vision row
#define CHUNK (HW_ / VCH)   // 4096 floats per chunk

// ---------------- K1: vision partial sums (the bandwidth-bound kernel) -------
// 192 rows (64 batches x 3 channels) x 64 chunks = 12288 blocks, 16B loads.
// Single-pass 201 MB stream > 192 MB L2 -> use non-temporal loads (TH=NT) so
// the sweep does not evict the small reusable operands (text_emb, weights).
__global__ void k_vision_partial(const float* __restrict__ vis,
                                 float* __restrict__ partials) {
    const int row   = blockIdx.x >> 6;   // 0..191
    const int chunk = blockIdx.x & 63;
    const v4f* p = (const v4f*)(vis + (size_t)row * HW_ + (size_t)chunk * CHUNK);
    const int t = threadIdx.x;
    float s = 0.0f;
#pragma unroll
    for (int i = 0; i < 4; ++i) {        // 4 x 16B = 16 floats/thread
        v4f v = __builtin_nontemporal_load(&p[t + i * 256]);
        s += v.x + v.y + v.z + v.w;
    }
    __shared__ float red[256];
    red[t] = s;
    __syncthreads();
    for (int off = 128; off > 0; off >>= 1) {
        if (t < off) red[t] += red[t + off];
        __syncthreads();
    }
    if (t == 0) partials[blockIdx.x] = red[0];
}

// ---------------- K2: finalize vision means ---------------------------------
__global__ void k_vision_final(const float* __restrict__ partials,
                               float* __restrict__ vmean) {
    const int r = threadIdx.x;
    if (r < 192) {
        float s = 0.0f;
        for (int c = 0; c < VCH; ++c) s += partials[r * VCH + c];
        vmean[r] = s * (1.0f / (float)HW_);
    }
}

// ---------------- K3: audio means (single-pass stream -> NT loads) ----------
__global__ void k_audio_mean(const float* __restrict__ audio,
                             float* __restrict__ amean) {
    const int b = blockIdx.x, t = threadIdx.x;
    const float* p = audio + (size_t)b * T_;
    float s = 0.0f;
    for (int i = t; i < T_; i += 256) s += __builtin_nontemporal_load(&p[i]);
    __shared__ float red[256];
    red[t] = s;
    __syncthreads();
    for (int off = 128; off > 0; off >>= 1) {
        if (t < off) red[t] += red[t + off];
        __syncthreads();
    }
    if (t == 0) amean[b] = red[0] * (1.0f / (float)T_);
}

// ---------------- K4: text byte histogram (gather-mean == hist @ emb) -------
__global__ void k_text_hist(const int* __restrict__ bytes,
                            float* __restrict__ hist) {
    __shared__ int h[256];
    const int b = blockIdx.x, t = threadIdx.x;
    h[t] = 0;
    __syncthreads();
    const int* p = bytes + (size_t)b * L_;
    for (int i = t; i < L_; i += 256)
        atomicAdd(&h[__builtin_nontemporal_load(&p[i]) & 255], 1);
    __syncthreads();
    hist[b * 256 + t] = (float)h[t] * (1.0f / (float)L_);
}

// ---------------- K5: tiny vision MLP stages 1-2 + audio broadcast ----------
__global__ void k_small(const float* __restrict__ vmean, const float* __restrict__ amean,
                        const float* __restrict__ Wv1, const float* __restrict__ bv1,
                        const float* __restrict__ Wv2, const float* __restrict__ bv2,
                        const float* __restrict__ Wa,  const float* __restrict__ ba,
                        float* __restrict__ h2out, float* __restrict__ aout) {
    __shared__ float h1[64 * 16];
    const int t = threadIdx.x;
    for (int i = t; i < 64 * 16; i += 256) {      // silu(v @ Wv1^T + bv1)
        const int b = i >> 4, j = i & 15;
        float z = bv1[j];
        for (int c = 0; c < 3; ++c) z += vmean[b * 3 + c] * Wv1[j * 3 + c];
        h1[i] = z / (1.0f + expf(-z));
    }
    __syncthreads();
    for (int i = t; i < 64 * 64; i += 256) {      // silu(h1 @ Wv2^T + bv2)
        const int b = i >> 6, j = i & 63;
        float z = bv2[j];
        for (int c = 0; c < 16; ++c) z += h1[b * 16 + c] * Wv2[j * 16 + c];
        h2out[i] = z / (1.0f + expf(-z));
    }
    for (int i = t; i < 64 * 256; i += 256) {     // a = mean(audio)*Wa + ba
        const int b = i >> 8, d = i & 255;
        aout[i] = amean[b] * Wa[d] + ba[d];
    }
}

// ---------------- WMMA f32 GEMM: D = epilogue(A @ B (+bias)) ----------------
// EP: 0 = plain store, 1 = +bias, 2 = add + tanh(z + bias)   BT: B transposed
template <int EP, bool BT>
__global__ void k_wmma_gemm(const float* __restrict__ A, int lda,
                            const float* __restrict__ Bm, int ldb,
                            const float* __restrict__ bias,
                            const float* __restrict__ addm, int ldadd,
                            float* __restrict__ D, int ldd,
                            int M, int N, int K) {
    const int lane   = threadIdx.x & 31;
    const int wave   = threadIdx.x >> 5;
    const int nWaves = blockDim.x >> 5;
    const int half   = lane >> 4;   // 0: K={0,1} ; 1: K={2,3}
    const int l16    = lane & 15;
    const int mTiles = M >> 4, nTiles = N >> 4;
    const int total  = mTiles * nTiles;

    for (int tile = wave; tile < total; tile += nWaves) {
        const int tm = (tile / nTiles) << 4;
        const int tn = (tile % nTiles) << 4;
        v8f acc = {0.f, 0.f, 0.f, 0.f, 0.f, 0.f, 0.f, 0.f};
        const float* arow = A + (size_t)(tm + l16) * lda + 2 * half;  // A 16x4 frag
        const int col = tn + l16;
        for (int k = 0; k < K; k += 4) {
            v2f a; a.x = arow[k]; a.y = arow[k + 1];
            v2f b;
            const int kk = k + 2 * half;
            if (BT) {  // B[k][n] = Bm[n*ldb + k]
                const float* bp = Bm + (size_t)col * ldb + kk;
                b.x = bp[0]; b.y = bp[1];
            } else {   // B[k][n] = Bm[k*ldb + n]
                b.x = Bm[(size_t)kk * ldb + col];
                b.y = Bm[(size_t)(kk + 1) * ldb + col];
            }
            acc = __builtin_amdgcn_wmma_f32_16x16x4_f32(
                false, a, false, b, (short)0, acc, false, false);
        }
#pragma unroll
        for (int r = 0; r < 8; ++r) {   // C/D: VGPR r -> row r (lanes<16) / r+8
            const int row = tm + r + 8 * half;
            float z = acc[r];
            if (EP >= 1) z += bias[col];
            if (EP == 2) z = addm[(size_t)row * ldadd + col] + tanhf(z);
            D[(size_t)row * ldd + col] = z;
        }
    }
}

// ---------------- K8: inner products -> sign(gamma) -> combined -------------
__global__ void k_gamma(const float* __restrict__ v, const float* __restrict__ t,
                        const float* __restrict__ a, float* __restrict__ comb) {
    const int b = blockIdx.x, d = threadIdx.x;
    const float vv = v[b * 256 + d], tt = t[b * 256 + d], aa = a[b * 256 + d];
    __shared__ float r0[256], r1[256], r2[256];
    r0[d] = vv * tt; r1[d] = tt * aa; r2[d] = aa * vv;
    __syncthreads();
    for (int off = 128; off > 0; off >>= 1) {
        if (d < off) { r0[d] += r0[d + off]; r1[d] += r1[d + off]; r2[d] += r2[d + off]; }
        __syncthreads();
    }
    const float prod = r0[0] * r1[0] * r2[0];
    // angle(x) for real x: 0 if x >= 0 else pi  ->  cos(angle) = +/-1
    const float c = (prod < 0.0f) ? -1.0f : 1.0f;
    comb[b * 256 + d] = (vv + tt + aa) * c;
}

// ---------------- K11: L1 normalize + write outputs (eta == 0 always) -------
__global__ void k_norm(const float* __restrict__ y1, const float* __restrict__ y2,
                       float* __restrict__ out) {
    const int b = blockIdx.x, d = threadIdx.x;
    const float m = (d < 128) ? y1[b * 128 + d] : y2[b * 128 + (d - 128)];
    __shared__ float red[256];
    red[d] = fabsf(m);
    __syncthreads();
    for (int off = 128; off > 0; off >>= 1) {
        if (d < off) red[d] += red[d + off];
        __syncthreads();
    }
    const float den = fmaxf(red[0], 1e-12f);
    out[b * 256 + d] = m / den;
    // det(I + m m^T) = 1 + ||m||^2 > 0  ->  sign = +1 -> angle = 0 -> eta = 0
    if (d == 0) out[64 * 256 + b] = 0.0f;
}

extern "C" void kernel_launch(void* const* d_in, const int* in_sizes, int n_in,
                              void* d_out, int out_size, void* d_ws, size_t ws_size,
                              hipStream_t stream) {
    const float* vis      = (const float*)d_in[0];
    const float* audio    = (const float*)d_in[1];
    const int*   tbytes   = (const int*)  d_in[2];
    const float* Wv1      = (const float*)d_in[3];
    const float* bv1      = (const float*)d_in[4];
    const float* Wv2      = (const float*)d_in[5];
    const float* bv2      = (const float*)d_in[6];
    const float* Wv3      = (const float*)d_in[7];
    const float* bv3      = (const float*)d_in[8];
    const float* text_emb = (const float*)d_in[9];
    const float* Wa       = (const float*)d_in[10];
    const float* ba       = (const float*)d_in[11];
    const float* Wf       = (const float*)d_in[12];
    const float* bf       = (const float*)d_in[13];
    const float* Wg       = (const float*)d_in[14];
    const float* bg       = (const float*)d_in[15];
    float* out = (float*)d_out;

    float* ws       = (float*)d_ws;
    float* partials = ws;                  // 192*64 = 12288
    float* vmean    = partials + 12288;    // 192
    float* amean    = vmean + 192;         // 64
    float* hist     = amean + 64;          // 64*256
    float* tmat     = hist + 16384;        // 64*256
    float* h2       = tmat + 16384;        // 64*64
    float* amat     = h2 + 4096;           // 64*256
    float* vmat     = amat + 16384;        // 64*256
    float* comb     = vmat + 16384;        // 64*256
    float* y1       = comb + 16384;        // 64*128
    float* y2       = y1 + 8192;           // 64*128  (total ~460 KB)

    // Stage 1: bulk reductions (bandwidth-bound part)
    k_vision_partial<<<192 * VCH, 256, 0, stream>>>(vis, partials);
    k_vision_final  <<<1, 256, 0, stream>>>(partials, vmean);
    k_audio_mean    <<<B_, 256, 0, stream>>>(audio, amean);
    k_text_hist     <<<B_, 256, 0, stream>>>(tbytes, hist);

    // Stage 2: tiny MLP + audio broadcast
    k_small<<<1, 256, 0, stream>>>(vmean, amean, Wv1, bv1, Wv2, bv2, Wa, ba, h2, amat);

    // Stage 3: WMMA GEMMs (f32 16x16x4)
    // t = hist @ text_emb           (M=64, N=256, K=256), B row-major
    k_wmma_gemm<0, false><<<1, 256, 0, stream>>>(hist, 256, text_emb, 256,
                                                 nullptr, nullptr, 0,
                                                 tmat, 256, 64, 256, 256);
    // v = h2 @ Wv3^T + bv3          (M=64, N=256, K=64), B transposed
    k_wmma_gemm<1, true><<<1, 256, 0, stream>>>(h2, 64, Wv3, 64,
                                                bv3, nullptr, 0,
                                                vmat, 256, 64, 256, 64);

    // Stage 4: phase sign + combined
    k_gamma<<<B_, 256, 0, stream>>>(vmat, tmat, amat, comb);

    // Stage 5: coupling flow (WMMA + tanh epilogue)
    // y1 = x1 + tanh(x2 @ Wf^T + bf) ; x1 = comb[:, :128], x2 = comb[:, 128:]
    k_wmma_gemm<2, true><<<1, 256, 0, stream>>>(comb + 128, 256, Wf, 128,
                                                bf, comb, 256,
                                                y1, 128, 64, 128, 128);
    // y2 = x2 + tanh(y1 @ Wg^T + bg)
    k_wmma_gemm<2, true><<<1, 256, 0, stream>>>(y1, 128, Wg, 128,
                                                bg, comb + 128, 256,
                                                y2, 128, 64, 128, 128);

    // Stage 6: L1 normalize, write m and eta
    k_norm<<<B_, 256, 0, stream>>>(y1, y2, out);
}